// ChebNet_84121229460235
// MI455X (gfx1250) — compile-verified
//
#include <hip/hip_runtime.h>

typedef float v2f __attribute__((ext_vector_type(2)));
typedef float v8f __attribute__((ext_vector_type(8)));

// ---------------- elementwise / setup kernels ----------------

__global__ void k_zero(float* __restrict__ p, int n) {
  int i = blockIdx.x * blockDim.x + threadIdx.x;
  if (i < n) p[i] = 0.0f;
}

__global__ void k_deg(const int* __restrict__ dst, float* __restrict__ deg, int E) {
  int e = blockIdx.x * blockDim.x + threadIdx.x;
  if (e < E) atomicAdd(&deg[dst[e]], 1.0f);
}

__global__ void k_norm(float* __restrict__ d, int N) {
  int i = blockIdx.x * blockDim.x + threadIdx.x;
  if (i < N) {
    float v = d[i];
    v = v < 1.0f ? 1.0f : v;
    d[i] = rsqrtf(v);
  }
}

// y[dst] += h[src] * norm[src]; 16 threads per edge, float4 per thread (D=64).
// All buffers are L2-resident (192MB L2 >> ~65MB working set), so these are L2
// gathers + L2 fp32 atomics — the true bottleneck of this workload.
__global__ void k_scatter(const float* __restrict__ h, const float* __restrict__ norm,
                          const int* __restrict__ src, const int* __restrict__ dst,
                          float* __restrict__ y, int E) {
  long long tid = (long long)blockIdx.x * blockDim.x + threadIdx.x;
  if (tid >= (long long)E * 16) return;
  int e = (int)(tid >> 4);
  int f = ((int)tid & 15) << 2;
  int s = src[e];
  int d = dst[e];
  float ns = norm[s];
  const float4 hv = *(const float4*)(h + (long long)s * 64 + f);
  float* yp = y + (long long)d * 64 + f;
  atomicAdd(yp + 0, hv.x * ns);
  atomicAdd(yp + 1, hv.y * ns);
  atomicAdd(yp + 2, hv.z * ns);
  atomicAdd(yp + 3, hv.w * ns);
}

// T1 = -(y * norm)   (T1 = -anorm(T0))
__global__ void k_finishT1(const float* __restrict__ y, const float* __restrict__ norm,
                           float* __restrict__ T1, int nTot) {
  int i = blockIdx.x * blockDim.x + threadIdx.x;
  if (i < nTot) T1[i] = -y[i] * norm[i >> 6];
}

// T2 = -2*(y*norm) - T0, written in place over y.
__global__ void k_finishT2(float* __restrict__ y, const float* __restrict__ norm,
                           const float* __restrict__ T0, int nTot) {
  int i = blockIdx.x * blockDim.x + threadIdx.x;
  if (i < nTot) y[i] = -2.0f * y[i] * norm[i >> 6] - T0[i];
}

// ---------------- WMMA GEMM: out = relu(T0@W0 + T1@W1 + T2@W2 + b) -------------
// Register-blocked: one wave32 owns a 32-row x 64-col output block
// (2 row tiles x 4 col tiles = 8 accumulators). Per k-step: 2 A-loads (b64),
// 8 B-loads shared by both row tiles, 8 x V_WMMA_F32_16X16X4_F32 (exact fp32).
//
// A (16x4 f32) layout: lane m = lane&15, v2f = {K=2*(lane>>4), K=2*(lane>>4)+1}
// B (4x16 f32) layout: lane n = lane&15, same K striping
// C/D (16x16 f32):     VGPR r -> row r + 8*(lane>>4), col n

template <bool TWO>
__device__ __forceinline__ void cheb_tile32(
    const float* __restrict__ T, const float* __restrict__ Wt,
    long long arow0, int boff, v8f acc0[4], v8f acc1[4]) {
#pragma unroll
  for (int kt = 0; kt < 64; kt += 4) {
    v2f bt[4];
#pragma unroll
    for (int c = 0; c < 4; ++c) {
      const float* bp = Wt + kt * 64 + boff + c * 16;
      bt[c].x = bp[0];
      bt[c].y = bp[64];
    }
    v2f a0 = *(const v2f*)(T + arow0 + kt);
#pragma unroll
    for (int c = 0; c < 4; ++c)
      acc0[c] = __builtin_amdgcn_wmma_f32_16x16x4_f32(false, a0, false, bt[c],
                                                      (short)0, acc0[c], false, false);
    if (TWO) {
      v2f a1 = *(const v2f*)(T + arow0 + 16 * 64 + kt);
#pragma unroll
      for (int c = 0; c < 4; ++c)
        acc1[c] = __builtin_amdgcn_wmma_f32_16x16x4_f32(false, a1, false, bt[c],
                                                        (short)0, acc1[c], false, false);
    }
  }
}

__global__ __launch_bounds__(256) void k_cheb_gemm(
    const float* __restrict__ T0, const float* __restrict__ T1,
    const float* __restrict__ T2, const float* __restrict__ W,
    const float* __restrict__ bias, float* __restrict__ out, int N) {
  const int lane = threadIdx.x & 31;
  const int wave = blockIdx.x * 8 + (threadIdx.x >> 5);
  const int r0 = wave * 32;
  if (r0 >= N) return;                       // wave-uniform: EXEC stays all-1s
  const bool two = (r0 + 32) <= N;           // second 16-row tile valid?

  const int m = lane & 15;                   // output row in tile / B column
  const int half = lane >> 4;                // K-pair selector
  const long long arow0 = (long long)(r0 + m) * 64 + (half << 1);
  const int boff = (half << 1) * 64 + m;

  v8f acc0[4] = {};
  v8f acc1[4] = {};
  if (two) {
    cheb_tile32<true>(T0, W, arow0, boff, acc0, acc1);
    cheb_tile32<true>(T1, W + 4096, arow0, boff, acc0, acc1);
    cheb_tile32<true>(T2, W + 8192, arow0, boff, acc0, acc1);
  } else {
    cheb_tile32<false>(T0, W, arow0, boff, acc0, acc1);
    cheb_tile32<false>(T1, W + 4096, arow0, boff, acc0, acc1);
    cheb_tile32<false>(T2, W + 8192, arow0, boff, acc0, acc1);
  }

  // epilogue: bias + relu, scatter accumulators per the C/D layout
  const int rowBase = r0 + (half << 3);
#pragma unroll
  for (int c = 0; c < 4; ++c) {
    const float bv = bias[c * 16 + m];
    float* op = out + (long long)rowBase * 64 + c * 16 + m;
#pragma unroll
    for (int r = 0; r < 8; ++r) {
      float v = acc0[c][r] + bv;
      op[(long long)r * 64] = v > 0.0f ? v : 0.0f;
    }
    if (two) {
      float* op1 = op + 16 * 64;
#pragma unroll
      for (int r = 0; r < 8; ++r) {
        float v = acc1[c][r] + bv;
        op1[(long long)r * 64] = v > 0.0f ? v : 0.0f;
      }
    }
  }
}

// ---------------- head: logits = h @ pw + pb ----------------
__global__ void k_head(const float* __restrict__ h, const float* __restrict__ pw,
                       const float* __restrict__ pb, float* __restrict__ out, int N) {
  int i = blockIdx.x * blockDim.x + threadIdx.x;
  if (i < N) {
    const float* hp = h + (long long)i * 64;
    float s = 0.0f;
#pragma unroll
    for (int j = 0; j < 64; ++j) s = fmaf(hp[j], pw[j], s);
    out[i] = s + pb[0];
  }
}

// ---------------- launcher ----------------
extern "C" void kernel_launch(void* const* d_in, const int* in_sizes, int n_in,
                              void* d_out, int out_size, void* d_ws, size_t ws_size,
                              hipStream_t stream) {
  const float* features = (const float*)d_in[0];
  const int*   src      = (const int*)d_in[1];
  const int*   dst      = (const int*)d_in[2];
  const float* W        = (const float*)d_in[3];
  const float* b        = (const float*)d_in[4];
  const float* pw       = (const float*)d_in[5];
  const float* pb       = (const float*)d_in[6];

  const int N = in_sizes[0] / 64;       // 50000 (divisible by 16)
  const int E = in_sizes[1];            // 800000
  const int Lnum = in_sizes[4] / 64;    // 2 layers

  // workspace layout (floats): norm[N] | y[N*64] | T1[N*64] | h1[N*64] | h2[N*64]
  const size_t NF = (size_t)N * 64;
  float* ws   = (float*)d_ws;
  float* norm = ws;
  float* ybuf = norm + N;
  float* T1   = ybuf + NF;
  float* h1   = T1 + NF;
  float* h2   = h1 + NF;

  const int thr = 256;
  const int nBlkN  = (N + thr - 1) / thr;
  const int nBlkE  = (E + thr - 1) / thr;
  const int nBlkNF = (int)((NF + thr - 1) / thr);
  const long long scThreads = (long long)E * 16;
  const int nBlkSc = (int)((scThreads + thr - 1) / thr);
  const int nWaves = (N + 31) / 32;          // one wave per 32-row block
  const int nBlkMM = (nWaves + 7) / 8;       // 8 waves per 256-thread block

  // degree -> norm = rsqrt(max(deg,1)) (in place)
  k_zero<<<nBlkN, thr, 0, stream>>>(norm, N);
  k_deg<<<nBlkE, thr, 0, stream>>>(dst, norm, E);
  k_norm<<<nBlkN, thr, 0, stream>>>(norm, N);

  const float* hin = features;
  float* houts[2] = {h1, h2};
  for (int l = 0; l < Lnum; ++l) {
    const float* Wl = W + (size_t)l * 3 * 64 * 64;
    const float* bl = b + (size_t)l * 64;
    float* hout = houts[l & 1];

    // T1 = -anorm(T0)
    k_zero<<<nBlkNF, thr, 0, stream>>>(ybuf, (int)NF);
    k_scatter<<<nBlkSc, thr, 0, stream>>>(hin, norm, src, dst, ybuf, E);
    k_finishT1<<<nBlkNF, thr, 0, stream>>>(ybuf, norm, T1, (int)NF);

    // T2 = -2*anorm(T1) - T0   (in place over ybuf)
    k_zero<<<nBlkNF, thr, 0, stream>>>(ybuf, (int)NF);
    k_scatter<<<nBlkSc, thr, 0, stream>>>(T1, norm, src, dst, ybuf, E);
    k_finishT2<<<nBlkNF, thr, 0, stream>>>(ybuf, norm, hin, (int)NF);

    // out = relu(T0@W0 + T1@W1 + T2@W2 + b) via register-blocked f32 WMMA
    k_cheb_gemm<<<nBlkMM, 256, 0, stream>>>(hin, T1, ybuf, Wl, bl, hout, N);
    hin = hout;
  }

  k_head<<<nBlkN, thr, 0, stream>>>(hin, pw, pb, (float*)d_out, N);
}